// ZEmb_53764400611469
// MI455X (gfx1250) — compile-verified
//
#include <hip/hip_runtime.h>

typedef __attribute__((ext_vector_type(2))) float v2f;
typedef __attribute__((ext_vector_type(8))) float v8f;

#define NPTS 32        // G
#define NN   8192      // N
#define GPB  128       // groups per block
#define NBIN 36        // 6*6 bins
#define KROW 144       // 36 bins * 4 features  (GEMM K)
#define KPAD 148       // padded row stride (bank spread, keeps 8B align)
#define MROWS 48       // 36 outputs padded to 3 WMMA M-tiles
#define OPAD 37

__launch_bounds__(256)
__global__ void zemb_kernel(const float* __restrict__ dp,
                            const float* __restrict__ cw,
                            const float* __restrict__ cb,
                            float* __restrict__ out) {
  __shared__ __align__(16) float Ml[MROWS * KPAD];   // conv as dense 48x144 map
  __shared__ __align__(16) float FQ[GPB * KPAD];     // per-group freq grids (B matrix)
  __shared__ __align__(16) float OT[GPB * OPAD];     // per-group conv outputs
  __shared__ unsigned char BIN[GPB * NPTS];          // per-point bin index

  const int tid  = threadIdx.x;
  const int lane = tid & 31;
  const int wave = tid >> 5;
  const long long grp_base = (long long)blockIdx.x * GPB;

  // ---- Build M from conv_w: out[oh][ow] += w[c][kh][kw]*freq[ih][iw][c]
  // ih = oh+kh-2 (zero-pad H), iw = (ow+kw-2) mod 6 (circular W) -> at most
  // one (kh,kw) per (o, f). Rows 36..47 zero-padded for the 3rd WMMA tile.
  for (int e = tid; e < MROWS * KROW; e += 256) {
    int o = e / KROW, f = e % KROW;
    float val = 0.f;
    if (o < NBIN) {
      int oh = o / 6, ow = o % 6;
      int c = f & 3, bin = f >> 2;
      int ih = bin / 6, iw = bin % 6;
      int kh = ih - oh + 2;
      int kw = (iw - ow + 8) % 6;
      if (kh >= 0 && kh < 5 && kw < 5) val = cw[c * 25 + kh * 5 + kw];
    }
    Ml[o * KPAD + f] = val;
  }
  // ---- Zero FREQ
  for (int e = tid * 4; e < GPB * KPAD; e += 1024)
    *(float4*)&FQ[e] = make_float4(0.f, 0.f, 0.f, 0.f);
  __syncthreads();

  // ---- Phase 1: wave-per-group (lane = point). 16 groups per wave.
  for (int t = 0; t < 16; ++t) {
    const int gl = wave * 16 + t;
    const long long grp = grp_base + gl;
    const int b = (int)(grp >> 13);          // / N
    const int n = (int)(grp & (NN - 1));
    const float* p = dp + (long long)b * 3 * NN * NPTS + (long long)n * NPTS + lane;
    const float x = p[0];
    const float y = p[NN * NPTS];
    const float z = p[2 * NN * NPTS];

    // argmax |dp|^2 across wave (first-max tie-break like jnp.argmax)
    float mv = x * x + y * y + z * z;
    int mi = lane;
    for (int off = 16; off >= 1; off >>= 1) {
      float ov = __shfl_xor(mv, off, 32);
      int oi = __shfl_xor(mi, off, 32);
      if (ov > mv || (ov == mv && oi < mi)) { mv = ov; mi = oi; }
    }
    float dx = __shfl(x, mi, 32), dy = __shfl(y, mi, 32), dz = __shfl(z, mi, 32);

    // normalize (1e-12), then rodrigues re-normalize (1e-8)
    float dn = sqrtf(dx * dx + dy * dy + dz * dz);
    float inv = 1.f / (dn + 1e-12f);
    float ax = dx * inv, ay = dy * inv, az = dz * inv;
    float an = sqrtf(ax * ax + ay * ay + az * az);
    float in2 = 1.f / (an + 1e-8f);
    ax *= in2; ay *= in2; az *= in2;

    // v = a x z = (ay, -ax, 0); s = |v|; c = az
    float s = sqrtf(ax * ax + ay * ay);
    float c = az;
    bool near_zero = (s < 1e-8f) && (c > 0.f);
    bool near_pi   = (s < 1e-8f) && (c < 0.f);
    float vx = ay, vy = -ax, vz = 0.f;
    if (near_pi) {
      float wx, wy, wz;
      if (fabsf(ax) < 0.9f) { wx = 0.f;  wy = az;  wz = -ay; } // a x (1,0,0)
      else                  { wx = -az;  wy = 0.f; wz = ax;  } // a x (0,1,0)
      float wn = sqrtf(wx * wx + wy * wy + wz * wz);
      float wi = 1.f / (wn + 1e-8f);
      vx = wx * wi; vy = wy * wi; vz = wz * wi;
    }
    float vn = sqrtf(vx * vx + vy * vy + vz * vz);
    float ki = 1.f / (vn + 1e-8f);
    float kx = vx * ki, ky = vy * ki, kz = vz * ki;
    float ic = 1.f - c;
    // R = I + s*K + (1-c)*K^2
    float r00 = 1.f - ic * (ky * ky + kz * kz);
    float r01 = -s * kz + ic * (kx * ky);
    float r02 =  s * ky + ic * (kx * kz);
    float r10 =  s * kz + ic * (kx * ky);
    float r11 = 1.f - ic * (kx * kx + kz * kz);
    float r12 = -s * kx + ic * (ky * kz);
    float r20 = -s * ky + ic * (kx * kz);
    float r21 =  s * kx + ic * (ky * kz);
    float r22 = 1.f - ic * (kx * kx + ky * ky);
    if (near_zero) { r00 = 1; r01 = 0; r02 = 0; r10 = 0; r11 = 1; r12 = 0; r20 = 0; r21 = 0; r22 = 1; }

    float px = r00 * x + r01 * y + r02 * z;
    float py = r10 * x + r11 * y + r12 * z;
    float pz = r20 * x + r21 * y + r22 * z;
    float rho = fmaxf(sqrtf(px * px + py * py + pz * pz), 1e-12f);

    // theta bin without arccos: count cos(theta) < cos(k*pi/6), k=1..5
    float cz = fminf(fmaxf(pz / rho, -1.f + 1e-6f), 1.f - 1e-6f);
    const float C1 = 0.8660254037844387f;
    int itb = (int)(cz < C1) + (int)(cz < 0.5f) + (int)(cz < 0.f) +
              (int)(cz < -0.5f) + (int)(cz < -C1);
    // phi bin without atan2: 60-degree sector tests
    const float SQ3 = 1.7320508075688772f;
    float xs = px, ys = py;
    int ipb = 0;
    if (py < 0.f) { xs = -px; ys = -py; ipb = 3; }
    if (ys < SQ3 * xs)        ipb += 0;
    else if (ys >= -SQ3 * xs) ipb += 1;
    else                      ipb += 2;

    float rm = rho;
    for (int off = 16; off >= 1; off >>= 1) rm = fmaxf(rm, __shfl_xor(rm, off, 32));
    float hf = fmaxf(0.5f - rho / (2.f * (rm + 1e-12f)), 0.f) + 0.5f;

    int bin = itb * 6 + ipb;
    BIN[gl * NPTS + lane] = (unsigned char)bin;
    float* fb = &FQ[gl * KPAD + bin * 4];
    atomicAdd(fb + 0, px);   // ds_add_f32
    atomicAdd(fb + 1, py);
    atomicAdd(fb + 2, pz);
    atomicAdd(fb + 3, hf);
  }
  __syncthreads();

  // ---- Phase 2: conv as GEMM on tensor units.
  // D(48x16) = Ml(48x144) x FQtile(144x16), f32 WMMA 16x16x4, 36 K-steps.
  // A layout (ISA 16x4 f32): lane half h holds K = {2h, 2h+1} for row M=lane%16.
  // B layout (mirrors C striping): lane half h holds rows K = {2h, 2h+1}, col N=lane%16.
  {
    const int ncol = lane & 15;
    const int hv = lane >> 4;
    const int gl0 = wave * 16 + ncol;
    v8f acc0 = {}; v8f acc1 = {}; v8f acc2 = {};
    for (int kb = 0; kb < 36; ++kb) {
      int f = kb * 4 + 2 * hv;
      v2f bf = *(const v2f*)&FQ[gl0 * KPAD + f];
      v2f a0 = *(const v2f*)&Ml[(ncol)      * KPAD + f];
      v2f a1 = *(const v2f*)&Ml[(16 + ncol) * KPAD + f];
      v2f a2 = *(const v2f*)&Ml[(32 + ncol) * KPAD + f];
      acc0 = __builtin_amdgcn_wmma_f32_16x16x4_f32(false, a0, false, bf, (short)0, acc0, false, false);
      acc1 = __builtin_amdgcn_wmma_f32_16x16x4_f32(false, a1, false, bf, (short)0, acc1, false, false);
      acc2 = __builtin_amdgcn_wmma_f32_16x16x4_f32(false, a2, false, bf, (short)0, acc2, false, false);
    }
    // D layout: VGPR r holds rows r (lanes 0-15) and r+8 (lanes 16-31), col = lane%16
    for (int r = 0; r < 8; ++r) {
      int o0 = r + 8 * hv;
      OT[gl0 * OPAD + o0] = acc0[r];
      OT[gl0 * OPAD + 16 + o0] = acc1[r];
      int o2 = 32 + o0;
      if (o2 < NBIN) OT[gl0 * OPAD + o2] = acc2[r];
    }
  }
  __syncthreads();

  // ---- Phase 3: gather conv output at each point's bin; coalesced stores.
  const float bias = cb[0];
  for (int i = 0; i < 16; ++i) {
    int pidx = i * 256 + tid;              // 0 .. 4095
    int gl = pidx >> 5;
    int g = pidx & 31;
    int bin = BIN[gl * NPTS + g];
    out[grp_base * NPTS + pidx] = OT[gl * OPAD + bin] + bias;
  }
}

extern "C" void kernel_launch(void* const* d_in, const int* in_sizes, int n_in,
                              void* d_out, int out_size, void* d_ws, size_t ws_size,
                              hipStream_t stream) {
  const float* dp = (const float*)d_in[0];   // (16,3,8192,32) f32
  const float* cw = (const float*)d_in[1];   // (1,4,5,5) f32
  const float* cb = (const float*)d_in[2];   // (1,) f32
  float* out = (float*)d_out;                // (16,1,8192,32) f32
  const long long n_groups = 16LL * 8192LL;  // B*N
  dim3 grid((unsigned)(n_groups / GPB));     // 1024 blocks
  zemb_kernel<<<grid, 256, 0, stream>>>(dp, cw, cb, out);
}